// SkipgramNegSamplingTT_76871324664462
// MI455X (gfx1250) — compile-verified
//
#include <hip/hip_runtime.h>

// ---------------------------------------------------------------------------
// TT skip-gram negative sampling on gfx1250 (MI455X).
// All three TT contractions run on V_WMMA_F32_16X16X4_F32:
//   stage1: [4x16 (pad 16)] @ [16x64]  -> res1[16x16]   (16 WMMA)
//   stage2: [16x16]         @ [16x64]  -> res2[64x16]   (16 WMMA, no waste)
//   stage3: [64x16]         @ [16x4 (pad 16)] -> emb    (16 WMMA)
// ---------------------------------------------------------------------------

typedef __attribute__((ext_vector_type(2))) float v2f;
typedef __attribute__((ext_vector_type(8))) float v8f;

#define NEG_K   20
#define SLOTS   22          // 1 center + 1 target + 20 negatives
#define WAVES   8
#define SCR_F   1088        // 64 rows * stride 17 (res1[16x17] aliases the front)

__device__ __forceinline__ float log_sigmoid(float x) {
    // stable: min(x,0) - log1p(exp(-|x|))
    return fminf(x, 0.0f) - log1pf(__expf(-fabsf(x)));
}

// One wave reconstructs one 256-d TT embedding row into `emb` (LDS).
// scr: per-wave LDS scratch, SCR_F floats.
__device__ __forceinline__ void tt_lookup_wave(
    int idx,
    const float* __restrict__ c0, const float* __restrict__ c1,
    const float* __restrict__ c2, const float* __restrict__ c3,
    float* __restrict__ scr, float* __restrict__ emb, int lane)
{
    // mixed-radix digits, factors (16,16,16,8), strides (2048,128,8,1)
    const int d0 = (idx >> 11) & 15;
    const int d1 = (idx >> 7)  & 15;
    const int d2 = (idx >> 3)  & 15;
    const int d3 =  idx        & 7;

    // fragment geometry (ISA 7.12.2, 32-bit fragments, wave32):
    //   A 16x4 : lane = M | (khalf<<4); vgpr0 K=k0+{0|2}, vgpr1 K=k0+{1|3}
    //   B 4x16 : lane = N | (khalf<<4); same K split
    //   C 16x16: vgpr i = row i (lanes 0-15) / row i+8 (lanes 16-31), col=lane&15
    const int am    = lane & 15;
    const int bn    = lane & 15;
    const int kb    = (lane < 16) ? 0 : 2;
    const int mhalf = (lane < 16) ? 0 : 8;

    const float* __restrict__ g1 = c1 + d1 * 64;   // G1[k,N] = g1[k*1024 + N]
    const float* __restrict__ g2 = c2 + d2 * 64;   // G2[k,N] = g2[k*1024 + N]

    // Warm the stage-2 slice (16 rows x 256B) in WGP$/L2: global_prefetch_b8.
    __builtin_prefetch(g2 + (lane & 15) * 1024 + (lane >> 4) * 32, 0, 1);

    // ---- stage 1 (WMMA): out1[16,64] = res0_pad[16,16-ish] @ G1[16,64]
    // res0 rows 0..3 valid; rows 4..15 zero (row index clamped to stay in-bounds).
    {
        const float* __restrict__ r0 = c0 + d0 * 64 + (am & 3) * 16;
        const bool mv = (am < 4);
        float a0s[4], a1s[4];
#pragma unroll
        for (int ks = 0; ks < 4; ++ks) {
            const int k = 4 * ks + kb;
            a0s[ks] = mv ? r0[k]     : 0.0f;
            a1s[ks] = mv ? r0[k + 1] : 0.0f;
        }
#pragma unroll
        for (int nt = 0; nt < 4; ++nt) {
            v8f C = {};
#pragma unroll
            for (int ks = 0; ks < 4; ++ks) {
                const int krow = 4 * ks + kb;
                v2f A; A[0] = a0s[ks]; A[1] = a1s[ks];
                v2f B;
                B[0] = g1[krow * 1024 + nt * 16 + bn];
                B[1] = g1[(krow + 1) * 1024 + nt * 16 + bn];
                C = __builtin_amdgcn_wmma_f32_16x16x4_f32(
                    false, A, false, B, (short)0, C, false, false);
            }
            // valid rows M=0..3 sit in vgprs 0..3 of lanes 0-15;
            // res1 row = 4*M + nt, col = bn  (stride-17 LDS, conflict-free)
            if (lane < 16) {
#pragma unroll
                for (int i = 0; i < 4; ++i)
                    scr[(4 * i + nt) * 17 + bn] = C[i];
            }
        }
    }
    asm volatile("s_wait_dscnt 0" ::: "memory");   // res1 visible wave-locally

    // ---- preload stage-2 A fragments (res1 in A 16x4 layout, all 4 K-steps)
    float a0[4], a1[4];
#pragma unroll
    for (int ks = 0; ks < 4; ++ks) {
        a0[ks] = scr[am * 17 + 4 * ks + kb];
        a1[ks] = scr[am * 17 + 4 * ks + kb + 1];
    }
    asm volatile("s_wait_dscnt 0" ::: "memory");   // A regs safe before res2 overwrite

    // ---- stage 2 (WMMA): out2[16,64] = res1[16,16] @ G2[16,64]  (no waste)
#pragma unroll
    for (int nt = 0; nt < 4; ++nt) {
        v8f C = {};
#pragma unroll
        for (int ks = 0; ks < 4; ++ks) {
            const int krow = 4 * ks + kb;
            v2f A; A[0] = a0[ks]; A[1] = a1[ks];
            v2f B;
            B[0] = g2[krow * 1024 + nt * 16 + bn];
            B[1] = g2[(krow + 1) * 1024 + nt * 16 + bn];
            C = __builtin_amdgcn_wmma_f32_16x16x4_f32(
                false, A, false, B, (short)0, C, false, false);
        }
        // C tile -> res2[64,16]: vgpr i = out2[M=i+mhalf, N=16*nt+bn];
        // res2 row = 4*M + nt, col = bn.
#pragma unroll
        for (int i = 0; i < 8; ++i) {
            const int row = 4 * (i + mhalf) + nt;
            scr[row * 17 + bn] = C[i];
        }
    }
    asm volatile("s_wait_dscnt 0" ::: "memory");   // res2 visible wave-locally

    // ---- stage 3 (WMMA): out3[64,4->pad16] = res2[64,16] @ G3_pad[16,16]
    // B cols 0..3 valid (col clamped to stay inside the 512-float core3).
    {
        const float* __restrict__ g3 = c3 + d3 * 4;   // g3[k*32 + m]
        const int  bcol = bn & 3;
        const bool nv   = (bn < 4);
        float b0s[4], b1s[4];
#pragma unroll
        for (int ks = 0; ks < 4; ++ks) {
            const int krow = 4 * ks + kb;
            b0s[ks] = nv ? g3[krow * 32 + bcol]       : 0.0f;
            b1s[ks] = nv ? g3[(krow + 1) * 32 + bcol] : 0.0f;
        }
#pragma unroll
        for (int mt = 0; mt < 4; ++mt) {
            v8f C = {};
#pragma unroll
            for (int ks = 0; ks < 4; ++ks) {
                const int krow = 4 * ks + kb;
                v2f A;
                A[0] = scr[(mt * 16 + am) * 17 + krow];
                A[1] = scr[(mt * 16 + am) * 17 + krow + 1];
                v2f B; B[0] = b0s[ks]; B[1] = b1s[ks];
                C = __builtin_amdgcn_wmma_f32_16x16x4_f32(
                    false, A, false, B, (short)0, C, false, false);
            }
            // vgpr i = out3[M=mt*16+i+mhalf, N=bn]; D index = M*4 + N (N<4)
            if (bn < 4) {
#pragma unroll
                for (int i = 0; i < 8; ++i)
                    emb[(mt * 16 + i + mhalf) * 4 + bn] = C[i];
            }
        }
    }
    // pin DS ordering before this wave's next lookup reuses scr
    asm volatile("s_wait_dscnt 0" ::: "memory");
}

__global__ __launch_bounds__(256)
void tt_sgns_kernel(const int* __restrict__ center,
                    const int* __restrict__ target,
                    const int* __restrict__ negw,
                    const float* __restrict__ v0, const float* __restrict__ v1,
                    const float* __restrict__ v2, const float* __restrict__ v3,
                    const float* __restrict__ u0, const float* __restrict__ u1,
                    const float* __restrict__ u2, const float* __restrict__ u3,
                    float* __restrict__ partial)
{
    __shared__ float emb[SLOTS][256];          // 22.0 KB
    __shared__ float scratch[WAVES][SCR_F];    // 34.0 KB
    __shared__ float red[2][256];              //  2.0 KB  (total 58 KB LDS)

    const int b    = blockIdx.x;
    const int lane = threadIdx.x & 31;
    const int wave = threadIdx.x >> 5;

    // 22 lookups round-robined over 8 waves (wave-uniform control flow: WMMA
    // executes with EXEC all-ones inside each wave).
    for (int slot = wave; slot < SLOTS; slot += WAVES) {
        int idx;
        const float *c0, *c1, *c2, *c3;
        if (slot == 0)      { idx = center[b];                c0=v0; c1=v1; c2=v2; c3=v3; }
        else if (slot == 1) { idx = target[b];                c0=u0; c1=u1; c2=u2; c3=u3; }
        else                { idx = negw[b*NEG_K + slot - 2]; c0=u0; c1=u1; c2=u2; c3=u3; }
        tt_lookup_wave(idx, c0, c1, c2, c3, scratch[wave], emb[slot], lane);
    }
    __syncthreads();

    // scores: pos = <target, center>; neg = -sum_k <u_emb(neg_k), center>
    const int d  = threadIdx.x;
    const float cv = emb[0][d];
    float p  = emb[1][d] * cv;
    float ns = 0.0f;
#pragma unroll
    for (int k = 0; k < NEG_K; ++k) ns += emb[2 + k][d] * cv;
    red[0][d] = p;
    red[1][d] = ns;
    __syncthreads();
#pragma unroll
    for (int s = 128; s > 0; s >>= 1) {        // deterministic tree reduction
        if (d < s) { red[0][d] += red[0][d + s]; red[1][d] += red[1][d + s]; }
        __syncthreads();
    }
    if (d == 0) {
        const float pos = red[0][0];
        const float neg = -red[1][0];
        partial[b] = log_sigmoid(pos) + log_sigmoid(neg);
    }
}

__global__ __launch_bounds__(256)
void tt_reduce_kernel(const float* __restrict__ partial, float* __restrict__ out, int n)
{
    __shared__ float sm[256];
    float s = 0.0f;
    for (int i = threadIdx.x; i < n; i += 256) s += partial[i];  // fixed order
    sm[threadIdx.x] = s;
    __syncthreads();
#pragma unroll
    for (int w = 128; w > 0; w >>= 1) {
        if (threadIdx.x < w) sm[threadIdx.x] += sm[threadIdx.x + w];
        __syncthreads();
    }
    if (threadIdx.x == 0) out[0] = -sm[0] / (float)n;
}

extern "C" void kernel_launch(void* const* d_in, const int* in_sizes, int n_in,
                              void* d_out, int out_size, void* d_ws, size_t ws_size,
                              hipStream_t stream)
{
    const int*   center = (const int*)d_in[0];
    const int*   target = (const int*)d_in[1];
    const int*   negw   = (const int*)d_in[2];
    const float* v0 = (const float*)d_in[3];
    const float* v1 = (const float*)d_in[4];
    const float* v2 = (const float*)d_in[5];
    const float* v3 = (const float*)d_in[6];
    const float* u0 = (const float*)d_in[7];
    const float* u1 = (const float*)d_in[8];
    const float* u2 = (const float*)d_in[9];
    const float* u3 = (const float*)d_in[10];

    const int B = in_sizes[0];                 // 8192 (center_words flat count)
    float* partial = (float*)d_ws;             // B floats of scratch

    tt_sgns_kernel<<<B, 256, 0, stream>>>(center, target, negw,
                                          v0, v1, v2, v3, u0, u1, u2, u3,
                                          partial);
    tt_reduce_kernel<<<1, 256, 0, stream>>>(partial, (float*)d_out, B);
}